// Sparsemax_17617955848439
// MI455X (gfx1250) — compile-verified
//
#include <hip/hip_runtime.h>

#ifndef __has_builtin
#define __has_builtin(x) 0
#endif

#define SM_N        512          // row length (last dim)
#define SM_WAVES    8            // waves per block (wave32) -> 8 rows per block
#define SM_THREADS  (SM_WAVES * 32)
#define SM_NEWTON   12           // Michelot/Newton iterations (exact on convergence)

// Types matching the async-to-LDS builtin signature on this toolchain:
// param0: int __vector(4) __device__*  (global, AS1)
// param1: LDS-side pointer (AS3)
typedef int v4i __attribute__((vector_size(16)));
typedef __attribute__((address_space(1))) v4i* gbl_v4i_ptr;
typedef __attribute__((address_space(3))) v4i* lds_v4i_ptr;

__device__ __forceinline__ float wave_reduce_max(float v) {
#pragma unroll
  for (int m = 16; m >= 1; m >>= 1) v = fmaxf(v, __shfl_xor(v, m, 32));
  return v;
}

__device__ __forceinline__ float wave_reduce_sum(float v) {
#pragma unroll
  for (int m = 16; m >= 1; m >>= 1) v += __shfl_xor(v, m, 32);
  return v;
}

__global__ __launch_bounds__(SM_THREADS) void sparsemax512_kernel(
    const float* __restrict__ in, float* __restrict__ out, int nrows) {
  __shared__ float lds[SM_WAVES * SM_N];   // 16 KB: one row per wave

  const int lane = threadIdx.x & 31;
  const int wave = threadIdx.x >> 5;
  const int row  = blockIdx.x * SM_WAVES + wave;
  if (row >= nrows) return;

  const float* __restrict__ grow = in + (size_t)row * SM_N;
  float* lrow = &lds[wave * SM_N];

  // ---- Stage row global -> LDS via gfx1250 async copies (ASYNCcnt path) ----
  // 4 chunks x 16B per lane = 64B/lane = 2KB/row, fully coalesced.
#if __has_builtin(__builtin_amdgcn_global_load_async_to_lds_b128)
#pragma unroll
  for (int c = 0; c < 4; ++c) {
    const int off = c * 128 + lane * 4;
    __builtin_amdgcn_global_load_async_to_lds_b128(
        (gbl_v4i_ptr)(grow + off),
        (lds_v4i_ptr)(lrow + off),
        0, 0);
  }
#else
#pragma unroll
  for (int c = 0; c < 4; ++c) {
    const int off = c * 128 + lane * 4;
    *(float4*)(lrow + off) = *(const float4*)(grow + off);
  }
#endif

#if __has_builtin(__builtin_amdgcn_s_wait_asynccnt)
  __builtin_amdgcn_s_wait_asynccnt(0);
#else
  asm volatile("s_wait_asynccnt 0" ::: "memory");
#endif
  // Wave only reads its own LDS region -> no workgroup barrier required.

  // ---- Pull 16 elements/lane into VGPRs (ds_load_b128) ----
  float z[16];
#pragma unroll
  for (int c = 0; c < 4; ++c) {
    float4 v = *(const float4*)(lrow + c * 128 + lane * 4);
    z[4 * c + 0] = v.x; z[4 * c + 1] = v.y;
    z[4 * c + 2] = v.z; z[4 * c + 3] = v.w;
  }

  // ---- Shift by row max (numerical stability; z_max becomes 0) ----
  float mx = z[0];
#pragma unroll
  for (int i = 1; i < 16; ++i) mx = fmaxf(mx, z[i]);
  mx = wave_reduce_max(mx);

  float s = 0.0f;
#pragma unroll
  for (int i = 0; i < 16; ++i) { z[i] -= mx; s += z[i]; }
  s = wave_reduce_sum(s);

  // ---- Newton/Michelot iteration for tau: tau <- (sum_{z>tau} z - 1)/count ----
  // f(tau)=sum(max(z-tau,0))-1 is convex piecewise-linear decreasing; this is
  // exact Newton and terminates at the exact root once the support stabilizes.
  // Since all z<=0 and z_max==0, every iterate has tau<0 => count>=1 always.
  float tau = (s - 1.0f) * (1.0f / (float)SM_N);
  for (int it = 0; it < SM_NEWTON; ++it) {
    float ss = 0.0f, cc = 0.0f;
#pragma unroll
    for (int i = 0; i < 16; ++i) {
      const bool a = z[i] > tau;
      ss += a ? z[i] : 0.0f;
      cc += a ? 1.0f : 0.0f;
    }
    ss = wave_reduce_sum(ss);
    cc = wave_reduce_sum(cc);
    tau = (ss - 1.0f) / cc;     // uniform across the wave -> no divergence
  }

  // ---- Emit p = max(z - tau, 0), coalesced b128 stores ----
  float* __restrict__ orow = out + (size_t)row * SM_N;
#pragma unroll
  for (int c = 0; c < 4; ++c) {
    float4 v;
    v.x = fmaxf(z[4 * c + 0] - tau, 0.0f);
    v.y = fmaxf(z[4 * c + 1] - tau, 0.0f);
    v.z = fmaxf(z[4 * c + 2] - tau, 0.0f);
    v.w = fmaxf(z[4 * c + 3] - tau, 0.0f);
    *(float4*)(orow + c * 128 + lane * 4) = v;
  }
}

extern "C" void kernel_launch(void* const* d_in, const int* in_sizes, int n_in,
                              void* d_out, int out_size, void* d_ws, size_t ws_size,
                              hipStream_t stream) {
  (void)n_in; (void)out_size; (void)d_ws; (void)ws_size;
  const float* x = (const float*)d_in[0];
  float* y = (float*)d_out;
  const int nrows  = in_sizes[0] / SM_N;               // 65536 for (64,1024,512)
  const int blocks = (nrows + SM_WAVES - 1) / SM_WAVES;
  sparsemax512_kernel<<<blocks, SM_THREADS, 0, stream>>>(x, y, nrows);
}